// GNNExample_59931973648439
// MI455X (gfx1250) — compile-verified
//
#include <hip/hip_runtime.h>

typedef __attribute__((ext_vector_type(2))) float v2f;
typedef __attribute__((ext_vector_type(8))) float v8f;

#define B_   128
#define N_   1024
#define F_   64
#define L_   256
#define E_   16384
#define IN_  (N_ * F_)   // 65536

// ---------------- utility ----------------
__global__ void zero_kernel(float* __restrict__ p, int n) {
  int i = blockIdx.x * blockDim.x + threadIdx.x;
  if (i < n) p[i] = 0.0f;
}

__global__ void deg_init_kernel(float* __restrict__ deg) {
  int i = blockIdx.x * blockDim.x + threadIdx.x;
  if (i < B_ * N_) deg[i] = 1.0f;   // self-loop contributes 1 to every node's degree
}

__global__ void deg_edge_kernel(const int* __restrict__ ei, float* __restrict__ deg) {
  int e = blockIdx.x * blockDim.x + threadIdx.x;
  int b = blockIdx.y;
  if (e < E_) {
    int d = ei[(size_t)b * 2 * E_ + E_ + e];   // dst row
    atomicAdd(&deg[b * N_ + d], 1.0f);
  }
}

__global__ void dinv_kernel(float* __restrict__ deg) {
  int i = blockIdx.x * blockDim.x + threadIdx.x;
  if (i < B_ * N_) deg[i] = rsqrtf(deg[i]);    // deg >= 1 always (self loops)
}

// ---------------- GCN linear: h[b] = x[b] @ W_conv  (1024x64 @ 64x64) ----------------
// one wave computes a 16x64 strip (all 4 N tiles) so the A fragment is reused 4x
__global__ __launch_bounds__(32)
void gcn_mm_kernel(const float* __restrict__ x, const float* __restrict__ Wc,
                   float* __restrict__ h) {
  const int b    = blockIdx.y;
  const int tm   = blockIdx.x;          // 0..63  (M tiles)
  const int lane = threadIdx.x;
  const int half = lane >> 4, l = lane & 15;
  const int M0 = tm * 16;
  const float* xb = x + (size_t)b * N_ * F_;

  v8f acc[4] = {};
  for (int k0 = 0; k0 < F_; k0 += 4) {
    v2f a;
    const float* ap = xb + (size_t)(M0 + l) * F_ + k0 + 2 * half;
    a.x = ap[0]; a.y = ap[1];
#pragma unroll
    for (int j = 0; j < 4; ++j) {
      v2f bb;
      bb.x = Wc[(size_t)(k0 + 2 * half)     * F_ + 16 * j + l];
      bb.y = Wc[(size_t)(k0 + 2 * half + 1) * F_ + 16 * j + l];
      acc[j] = __builtin_amdgcn_wmma_f32_16x16x4_f32(false, a, false, bb,
                                                     (short)0, acc[j], false, false);
    }
  }
  float* hb = h + (size_t)b * N_ * F_;
#pragma unroll
  for (int j = 0; j < 4; ++j)
#pragma unroll
    for (int r = 0; r < 8; ++r)
      hb[(size_t)(M0 + r + 8 * half) * F_ + 16 * j + l] = acc[j][r];
}

// ---------------- edge scatter: agg[dst] += h[src] * dinv[src]*dinv[dst] ----------------
__global__ __launch_bounds__(256)
void scatter_kernel(const int* __restrict__ ei, const float* __restrict__ h,
                    const float* __restrict__ dinv, float* __restrict__ agg) {
  const int b   = blockIdx.y;
  const int tid = threadIdx.x;         // 256 threads = 4 edges x 64 features
  const int el  = tid >> 6;
  const int f   = tid & 63;
  const int e   = blockIdx.x * 4 + el;
  const int s = ei[(size_t)b * 2 * E_ + e];
  const int d = ei[(size_t)b * 2 * E_ + E_ + e];
  const float norm = dinv[b * N_ + s] * dinv[b * N_ + d];
  atomicAdd(&agg[((size_t)b * N_ + d) * F_ + f],
            h[((size_t)b * N_ + s) * F_ + f] * norm);
}

// ---------------- finalize: xs = relu(agg + h*dinv^2 (self loop) + b_conv) ----------------
__global__ void gcn_finalize_kernel(const float* __restrict__ h,
                                    const float* __restrict__ dinv,
                                    const float* __restrict__ bconv,
                                    float* __restrict__ agg /* in/out -> xs */) {
  size_t i = (size_t)blockIdx.x * blockDim.x + threadIdx.x;
  if (i >= (size_t)B_ * N_ * F_) return;
  int    f  = (int)(i & 63);
  size_t bn = i >> 6;                  // b*N + n
  float  di = dinv[bn];
  float  v  = agg[i] + h[i] * di * di + bconv[f];
  agg[i] = v > 0.0f ? v : 0.0f;
}

// ---------------- encoder: enc = relu(xs @ W_enc + b_enc)  M=128 N=256 K=65536 ----------------
// 2x2 register tile per wave, split-K over 4 waves per block, LDS reduction
__global__ __launch_bounds__(128)
void encoder_kernel(const float* __restrict__ xs, const float* __restrict__ We,
                    const float* __restrict__ be, float* __restrict__ enc) {
  const int ng   = blockIdx.x;         // 0..7  (32-col groups)
  const int mg   = blockIdx.y;         // 0..3  (32-row groups)
  const int wave = threadIdx.x >> 5;
  const int lane = threadIdx.x & 31;
  const int half = lane >> 4, l = lane & 15;
  const int M0 = mg * 32, N0 = ng * 32;
  const int KS = IN_ / 4;              // 16384 per wave
  const int kb = wave * KS;

  v8f acc[2][2] = {};
  for (int k = 0; k < KS; k += 4) {
    const int kk = kb + k;
    v2f a[2], bv[2];
#pragma unroll
    for (int i = 0; i < 2; ++i) {
      const float* ap = xs + (size_t)(M0 + 16 * i + l) * IN_ + kk + 2 * half;
      a[i].x = ap[0]; a[i].y = ap[1];
    }
#pragma unroll
    for (int j = 0; j < 2; ++j) {
      bv[j].x = We[(size_t)(kk + 2 * half)     * L_ + N0 + 16 * j + l];
      bv[j].y = We[(size_t)(kk + 2 * half + 1) * L_ + N0 + 16 * j + l];
    }
#pragma unroll
    for (int i = 0; i < 2; ++i)
#pragma unroll
      for (int j = 0; j < 2; ++j)
        acc[i][j] = __builtin_amdgcn_wmma_f32_16x16x4_f32(false, a[i], false, bv[j],
                                                          (short)0, acc[i][j], false, false);
  }

  __shared__ float red[4][32][4][8];   // wave, lane, tile(ixj), r   = 16 KB
#pragma unroll
  for (int i = 0; i < 2; ++i)
#pragma unroll
    for (int j = 0; j < 2; ++j)
#pragma unroll
      for (int r = 0; r < 8; ++r)
        red[wave][lane][i * 2 + j][r] = acc[i][j][r];
  __syncthreads();

  if (wave == 0) {
#pragma unroll
    for (int i = 0; i < 2; ++i)
#pragma unroll
      for (int j = 0; j < 2; ++j) {
        const float bias = be[N0 + 16 * j + l];
#pragma unroll
        for (int r = 0; r < 8; ++r) {
          const int t = i * 2 + j;
          float s = red[0][lane][t][r] + red[1][lane][t][r]
                  + red[2][lane][t][r] + red[3][lane][t][r];
          s += bias;
          s = s > 0.0f ? s : 0.0f;
          enc[(size_t)(M0 + 16 * i + r + 8 * half) * L_ + N0 + 16 * j + l] = s;
        }
      }
  }
}

// ---------------- decoder: out = enc @ W_dec + b_dec   M=128 K=256 N=65536 ----------------
// one wave computes the full 128-row x 16-col strip: W_dec fragment loaded once, 8 WMMAs
__global__ __launch_bounds__(32)
void decoder_kernel(const float* __restrict__ enc, const float* __restrict__ Wd,
                    const float* __restrict__ bd, float* __restrict__ out) {
  const int ntile = blockIdx.x;        // 0..4095
  const int lane  = threadIdx.x;
  const int half  = lane >> 4, l = lane & 15;
  const int N0 = ntile * 16;

  v8f acc[8] = {};
  for (int k = 0; k < L_; k += 4) {
    v2f bb;
    bb.x = Wd[(size_t)(k + 2 * half)     * IN_ + N0 + l];
    bb.y = Wd[(size_t)(k + 2 * half + 1) * IN_ + N0 + l];
#pragma unroll
    for (int m = 0; m < 8; ++m) {
      v2f a;
      const float* ap = enc + (size_t)(16 * m + l) * L_ + k + 2 * half;
      a.x = ap[0]; a.y = ap[1];
      acc[m] = __builtin_amdgcn_wmma_f32_16x16x4_f32(false, a, false, bb,
                                                     (short)0, acc[m], false, false);
    }
  }
  const float bias = bd[N0 + l];
#pragma unroll
  for (int m = 0; m < 8; ++m)
#pragma unroll
    for (int r = 0; r < 8; ++r)
      out[(size_t)(16 * m + r + 8 * half) * IN_ + N0 + l] = acc[m][r] + bias;
}

// ---------------- launch ----------------
extern "C" void kernel_launch(void* const* d_in, const int* in_sizes, int n_in,
                              void* d_out, int out_size, void* d_ws, size_t ws_size,
                              hipStream_t stream) {
  const float* x  = (const float*)d_in[0];
  const int*   ei = (const int*)  d_in[1];
  const float* Wc = (const float*)d_in[2];
  const float* bc = (const float*)d_in[3];
  const float* We = (const float*)d_in[4];
  const float* be = (const float*)d_in[5];
  const float* Wd = (const float*)d_in[6];
  const float* bd = (const float*)d_in[7];
  float* out = (float*)d_out;

  // workspace layout (floats): h | agg/xs | dinv | enc   (~67.8 MB total)
  float* ws   = (float*)d_ws;
  float* h    = ws;                         // 8,388,608
  float* agg  = ws + (size_t)8388608;       // 8,388,608
  float* dinv = ws + (size_t)16777216;      //   131,072
  float* enc  = ws + (size_t)16908288;      //    32,768

  const int nAgg = B_ * N_ * F_;            // 8,388,608
  zero_kernel    <<<(nAgg + 255) / 256, 256, 0, stream>>>(agg, nAgg);
  deg_init_kernel<<<(B_ * N_ + 255) / 256, 256, 0, stream>>>(dinv);
  deg_edge_kernel<<<dim3(E_ / 256, B_), 256, 0, stream>>>(ei, dinv);
  dinv_kernel    <<<(B_ * N_ + 255) / 256, 256, 0, stream>>>(dinv);

  gcn_mm_kernel  <<<dim3(64, B_), 32, 0, stream>>>(x, Wc, h);
  scatter_kernel <<<dim3(E_ / 4, B_), 256, 0, stream>>>(ei, h, dinv, agg);
  gcn_finalize_kernel<<<(nAgg + 255) / 256, 256, 0, stream>>>(h, dinv, bc, agg);

  encoder_kernel <<<dim3(8, 4), 128, 0, stream>>>(agg, We, be, enc);
  decoder_kernel <<<dim3(IN_ / 16, 1), 32, 0, stream>>>(enc, Wd, bd, out);
}